// ConvSE3_68547678044850
// MI455X (gfx1250) — compile-verified
//
#include <hip/hip_runtime.h>

typedef float v2f __attribute__((ext_vector_type(2)));
typedef float v8f __attribute__((ext_vector_type(8)));
typedef unsigned int u32x4 __attribute__((ext_vector_type(4)));
typedef int i32x4 __attribute__((ext_vector_type(4)));
typedef int i32x8 __attribute__((ext_vector_type(8)));

#define E_TOT   4096
#define C_DIM   32
#define FREQ    44
#define SUMD    16
#define MID     32
#define ETILE   16
#define WAVES   16
#define TPB     (WAVES * 32)
#define W3_N    45056   // FREQ * C * C
#define BAS_ES  11264   // 16 * 44 * 16 floats per edge
#define BAS_IS  704     // 44 * 16 floats per i-row

static __device__ __forceinline__ v8f vzero8() {
    v8f r;
#pragma unroll
    for (int q = 0; q < 8; ++q) r[q] = 0.f;
    return r;
}

static __device__ __forceinline__ v8f wmma4(v2f a, v2f b, v8f c) {
    // V_WMMA_F32_16X16X4_F32: D = A(16x4) * B(4x16) + C(16x16), fp32 native
    return __builtin_amdgcn_wmma_f32_16x16x4_f32(false, a, false, b, (short)0, c,
                                                 false, false);
}

// ---------------------------------------------------------------------------
// Tensor Data Mover: async-load a 2D tile (16 rows x 16 contiguous fp32,
// row stride 704 floats) from global into LDS.  D# layout per CDNA5 ISA ch.8.
// ---------------------------------------------------------------------------
static __device__ __forceinline__ void tdm_load_tile16x16(unsigned lds_off,
                                                          unsigned long long gaddr) {
    u32x4 g0;
    g0[0] = 1u;                                            // count=1, user D#
    g0[1] = lds_off;                                       // LDS byte address
    g0[2] = (unsigned)(gaddr & 0xffffffffull);             // global_addr[31:0]
    g0[3] = (unsigned)((gaddr >> 32) & 0x1ffffffull)       // global_addr[56:32]
            | (2u << 30);                                  // type = 2 ("image")
    i32x8 g1;
    g1[0] = (2 << 16);        // workgroup_mask=0, data_size=2 (4 bytes)
    g1[1] = (16 << 16);       // tensor_dim0 = 16   (bits 79:48, low 16)
    g1[2] = (16 << 16);       // tensor_dim0 hi = 0, tensor_dim1 = 16 (low 16)
    g1[3] = (16 << 16);       // tensor_dim1 hi = 0, tile_dim0 = 16
    g1[4] = 16;               // tile_dim1 = 16, tile_dim2 = 0
    g1[5] = BAS_IS;           // tensor_dim0_stride = 704 (low 32)
    g1[6] = 0;                // stride hi, tensor_dim1_stride lo
    g1[7] = 0;
    i32x4 g2, g3;
#pragma unroll
    for (int q = 0; q < 4; ++q) { g2[q] = 0; g3[q] = 0; }
#if __has_include(<hip/amd_detail/amd_gfx1250_TDM.h>)
    i32x8 gz;
#pragma unroll
    for (int q = 0; q < 8; ++q) gz[q] = 0;
    __builtin_amdgcn_tensor_load_to_lds(g0, g1, g2, g3, gz, 0);   // clang-23 form
#else
    __builtin_amdgcn_tensor_load_to_lds(g0, g1, g2, g3, 0);       // ROCm 7.2 form
#endif
}

// concat([nf0,nf1,nf2,nf3], axis=-1)[n][c][i] with dims 1,3,5,7 (offsets 0,1,4,9)
static __device__ __forceinline__ float gather_feat(
    const float* __restrict__ nf0, const float* __restrict__ nf1,
    const float* __restrict__ nf2, const float* __restrict__ nf3,
    int n, int c, int i) {
    if (i < 1) return nf0[n * 32 + c];
    if (i < 4) return nf1[(n * 32 + c) * 3 + (i - 1)];
    if (i < 9) return nf2[(n * 32 + c) * 5 + (i - 4)];
    return nf3[(n * 32 + c) * 7 + (i - 9)];
}

// ---------------------------------------------------------------------------
// Kernel 1: radial MLP  h = relu(relu(inv @ w1 + b1) @ w2 + b2)   (E x 32)
// ---------------------------------------------------------------------------
__global__ void mlp_kernel(const float* __restrict__ ef, const float* __restrict__ w1,
                           const float* __restrict__ b1, const float* __restrict__ w2,
                           const float* __restrict__ b2, float* __restrict__ h) {
    int e = blockIdx.x * blockDim.x + threadIdx.x;
    if (e >= E_TOT) return;
    float x0 = ef[e * 2 + 0], x1 = ef[e * 2 + 1];
    float t[MID];
#pragma unroll
    for (int m = 0; m < MID; ++m) {
        float v = x0 * w1[m] + x1 * w1[MID + m] + b1[m];
        t[m] = v > 0.f ? v : 0.f;
    }
    for (int m2 = 0; m2 < MID; ++m2) {
        float acc = b2[m2];
#pragma unroll
        for (int m = 0; m < MID; ++m) acc += t[m] * w2[m * MID + m2];
        h[e * MID + m2] = acc > 0.f ? acc : 0.f;
    }
}

// ---------------------------------------------------------------------------
// Kernel 2: fused radial-GEMM + basis contraction + pairwise matmul + self
//           + atomic scatter.  One workgroup = 16 edges, one wave per edge.
//           basis tiles streamed by the Tensor Data Mover (double-buffered).
// ---------------------------------------------------------------------------
__global__ __launch_bounds__(TPB, 1)
void fused_kernel(const float* __restrict__ h, const float* __restrict__ basis,
                  const int* __restrict__ src, const int* __restrict__ dst,
                  const float* __restrict__ w3,
                  const float* __restrict__ nf0, const float* __restrict__ nf1,
                  const float* __restrict__ nf2, const float* __restrict__ nf3,
                  const float* __restrict__ k0, const float* __restrict__ k1,
                  const float* __restrict__ k2, const float* __restrict__ k3,
                  float* __restrict__ out) {
    __shared__ float Hs[ETILE][MID];              //  2 KB  H tile (16 edges x 32)
    __shared__ float Fs[ETILE][C_DIM][SUMD];      // 32 KB  gathered src features
    __shared__ float Rf[ETILE][C_DIM * C_DIM];    // 64 KB  radial slice [e][o*32+c]
    __shared__ float Tf[ETILE][C_DIM][SUMD];      // 32 KB  tmp slice / later nf[dst]
    __shared__ float Bf[2][ETILE][SUMD][SUMD];    // 32 KB  TDM basis tiles (dbl buf)

    const int tid  = threadIdx.x;
    const int wave = tid >> 5;       // = local edge index
    const int lane = tid & 31;
    const int l16  = lane & 15;
    const int lh   = lane >> 4;      // lane half
    const int e0   = blockIdx.x * ETILE;

    // ---- stage H tile (exactly 512 elements) and src features -------------
    {
        int eL = tid >> 5, m = tid & 31;
        Hs[eL][m] = h[(e0 + eL) * MID + m];
    }
    for (int idx = tid; idx < ETILE * C_DIM * SUMD; idx += TPB) {
        int eL = idx >> 9, c = (idx >> 4) & 31, i = idx & 15;
        int n = src[e0 + eL];
        Fs[eL][c][i] = gather_feat(nf0, nf1, nf2, nf3, n, c, i);
    }

    // ---- TDM prologue: stream basis tile for f=0 into buffer 0 ------------
    const unsigned long long basE =
        (unsigned long long)(uintptr_t)basis + (unsigned long long)(e0 + wave) * (BAS_ES * 4ull);
    tdm_load_tile16x16((unsigned)(uintptr_t)&Bf[0][wave][0][0], basE /* + 0*64 */);

    __syncthreads();

    // ---- hoisted A fragments ----------------------------------------------
    // radial GEMM A: rows = edge (M=l16), K = m (32 -> 8 chunks of 4)
    v2f aH[8];
#pragma unroll
    for (int kk = 0; kk < 8; ++kk) {
        aH[kk].x = Hs[l16][4 * kk + 2 * lh + 0];
        aH[kk].y = Hs[l16][4 * kk + 2 * lh + 1];
    }
    // Tf GEMM A: rows = c (2 m-tiles), K = i (16 -> 4 chunks of 4)
    v2f aF0[4], aF1[4];
#pragma unroll
    for (int kk = 0; kk < 4; ++kk) {
        aF0[kk].x = Fs[wave][l16][4 * kk + 2 * lh + 0];
        aF0[kk].y = Fs[wave][l16][4 * kk + 2 * lh + 1];
        aF1[kk].x = Fs[wave][16 + l16][4 * kk + 2 * lh + 0];
        aF1[kk].y = Fs[wave][16 + l16][4 * kk + 2 * lh + 1];
    }

    v8f accO0 = vzero8();   // out rows o=0..15
    v8f accO1 = vzero8();   // out rows o=16..31

    for (int f = 0; f < FREQ; ++f) {
        // ---- radial slice: Rf[e][o*32+c] = sum_m H[e][m] * w3[m][o*1408+c*44+f]
        // 64 n-tiles total, 4 per wave.  Overlaps the in-flight TDM transfer.
#pragma unroll
        for (int t4 = 0; t4 < 4; ++t4) {
            int ncol = (wave * 4 + t4) * 16 + l16;
            int o = ncol >> 5, c = ncol & 31;
            const float* w3col = w3 + (size_t)o * 1408 + (size_t)c * FREQ + f;
            __builtin_prefetch(w3col + 1, 0, 0);   // next-f column (L2 resident)
            v8f acc = vzero8();
#pragma unroll
            for (int kk = 0; kk < 8; ++kk) {
                int m0 = 4 * kk + 2 * lh;
                v2f b;
                b.x = w3col[(size_t)m0 * W3_N];
                b.y = w3col[(size_t)(m0 + 1) * W3_N];
                acc = wmma4(aH[kk], b, acc);
            }
#pragma unroll
            for (int v = 0; v < 8; ++v) Rf[v + 8 * lh][ncol] = acc[v];
        }

        // ---- kick TDM for f+1, then wait for tile f -------------------------
        if (f + 1 < FREQ) {
            tdm_load_tile16x16((unsigned)(uintptr_t)&Bf[(f + 1) & 1][wave][0][0],
                               basE + (unsigned long long)(f + 1) * 64ull);
            __builtin_amdgcn_s_wait_tensorcnt(1);   // tile f landed in LDS
        } else {
            __builtin_amdgcn_s_wait_tensorcnt(0);
        }

        // ---- Tf = feats(32x16) @ basis_f(16x16) from the TDM LDS tile -------
        {
            const int buf = f & 1;
            v8f t0 = vzero8(), t1 = vzero8();
#pragma unroll
            for (int kk = 0; kk < 4; ++kk) {
                int i0 = 4 * kk + 2 * lh;
                v2f b;
                b.x = Bf[buf][wave][i0][l16];
                b.y = Bf[buf][wave][i0 + 1][l16];
                t0 = wmma4(aF0[kk], b, t0);
                t1 = wmma4(aF1[kk], b, t1);
            }
#pragma unroll
            for (int v = 0; v < 8; ++v) {
                Tf[wave][v + 8 * lh][l16]      = t0[v];
                Tf[wave][16 + v + 8 * lh][l16] = t1[v];
            }
        }
        __syncthreads();

        // ---- out(32x16) += Rf_e(32x32) @ Tf(32x16) --------------------------
#pragma unroll
        for (int kk = 0; kk < 8; ++kk) {
            int cb = 4 * kk + 2 * lh;
            v2f b;
            b.x = Tf[wave][cb][l16];
            b.y = Tf[wave][cb + 1][l16];
            v2f a0, a1;
            a0.x = Rf[wave][l16 * 32 + cb];
            a0.y = Rf[wave][l16 * 32 + cb + 1];
            a1.x = Rf[wave][(16 + l16) * 32 + cb];
            a1.y = Rf[wave][(16 + l16) * 32 + cb + 1];
            accO0 = wmma4(a0, b, accO0);
            accO1 = wmma4(a1, b, accO1);
        }
        __syncthreads();   // Rf/Tf reused next iteration
    }

    // ---- self-interaction: stage node_feats[dst[e]] into Tf (reuse) --------
    for (int idx = tid; idx < ETILE * C_DIM * SUMD; idx += TPB) {
        int eL = idx >> 9, c = (idx >> 4) & 31, i = idx & 15;
        int n = dst[e0 + eL];
        Tf[eL][c][i] = gather_feat(nf0, nf1, nf2, nf3, n, c, i);
    }
    __syncthreads();

    const int de = dst[e0 + wave];
#pragma unroll
    for (int mt = 0; mt < 2; ++mt) {
#pragma unroll
        for (int v = 0; v < 8; ++v) {
            int o = mt * 16 + v + 8 * lh;
            int s = l16;
            const float* K = (s < 1) ? k0 : (s < 4) ? k1 : (s < 9) ? k2 : k3;
            float self = 0.f;
#pragma unroll
            for (int c = 0; c < 32; ++c) self += K[o * 32 + c] * Tf[wave][c][s];
            float val = (mt == 0 ? accO0[v] : accO1[v]) + self;
            atomicAdd(&out[(de * C_DIM + o) * SUMD + s], val);
        }
    }
}

extern "C" void kernel_launch(void* const* d_in, const int* in_sizes, int n_in,
                              void* d_out, int out_size, void* d_ws, size_t ws_size,
                              hipStream_t stream) {
    const float* nf0   = (const float*)d_in[0];
    const float* nf1   = (const float*)d_in[1];
    const float* nf2   = (const float*)d_in[2];
    const float* nf3   = (const float*)d_in[3];
    const float* ef    = (const float*)d_in[4];
    const float* basis = (const float*)d_in[5];
    const int*   src   = (const int*)d_in[6];
    const int*   dst   = (const int*)d_in[7];
    const float* w1    = (const float*)d_in[8];
    const float* b1    = (const float*)d_in[9];
    const float* w2    = (const float*)d_in[10];
    const float* b2    = (const float*)d_in[11];
    const float* w3    = (const float*)d_in[12];
    const float* k0    = (const float*)d_in[13];
    const float* k1    = (const float*)d_in[14];
    const float* k2    = (const float*)d_in[15];
    const float* k3    = (const float*)d_in[16];
    float* out = (float*)d_out;
    float* h   = (float*)d_ws;   // E*32 floats of scratch

    hipMemsetAsync(out, 0, (size_t)out_size * sizeof(float), stream);
    mlp_kernel<<<(E_TOT + 127) / 128, 128, 0, stream>>>(ef, w1, b1, w2, b2, h);
    fused_kernel<<<E_TOT / ETILE, TPB, 0, stream>>>(h, basis, src, dst, w3,
                                                    nf0, nf1, nf2, nf3,
                                                    k0, k1, k2, k3, out);
}